// LocallyConnected2dV1_65962107732544
// MI455X (gfx1250) — compile-verified
//
#include <hip/hip_runtime.h>
#include <hip/hip_bf16.h>
#include <stdint.h>

typedef float v2f __attribute__((ext_vector_type(2)));
typedef float v8f __attribute__((ext_vector_type(8)));

#define BATCH   64
#define HDIM    224
#define WDIM    224
#define KHN     7
#define KWN     7
#define STR     2
#define PADN    3
#define NOUT    112     // output is 112x112
#define JT      16      // j-tile width (N dim of WMMA)
#define BT      32      // batches per workgroup (2 waves x 16)
#define NK      40      // padded K extent: 10 wmma steps of K=4
#define NCOL    37      // valid staged columns: 2*15 + 7
#define XSTRIDE 42      // padded LDS row stride (floats); gcd(42,64)=2 -> few bank conflicts

// out[b, i, j] = sum_{kh,kk} xs[b, kh, kk] * Bband[kh][kk][n] ,  kk = 2n + kw
__global__ __launch_bounds__(64)
void lc2d_wmma_kernel(const float* __restrict__ x,
                      const float* __restrict__ w,
                      const float* __restrict__ bias,
                      float* __restrict__ out)
{
    __shared__ __align__(16) float xs[KHN * BT * XSTRIDE];  // staged input window (zero-padded)
    __shared__ __align__(16) float bb[KHN * JT * XSTRIDE];  // banded weights: [kh][n][kk]

    const int tid  = threadIdx.x;        // 0..63 (2 waves)
    const int lane = tid & 31;
    const int wv   = tid >> 5;           // wave id -> 16-batch tile within WG
    const int j0   = blockIdx.x * JT;    // 0,16,...,96
    const int i    = blockIdx.y;         // 0..111
    const int b0   = blockIdx.z * BT;    // 0 or 32

    // ---- 1) zero the x staging buffer (halo / padding / K-pad become zeros) ----
    for (int idx = tid; idx < KHN * BT * XSTRIDE; idx += 64)
        xs[idx] = 0.0f;

    // ---- 2) build banded B matrix in LDS: bb[kh][n][kk] = w[i, j0+n, kh, kk-2n] ----
    for (int idx = tid; idx < KHN * JT * XSTRIDE; idx += 64) {
        int kh  = idx / (JT * XSTRIDE);
        int rem = idx - kh * (JT * XSTRIDE);
        int n   = rem / XSTRIDE;
        int kk  = rem - n * XSTRIDE;
        int kw  = kk - 2 * n;
        float v = 0.0f;
        if (kw >= 0 && kw < KWN && kk < NK)
            v = w[(((size_t)i * NOUT + (j0 + n)) * KHN + kh) * KWN + kw];
        bb[idx] = v;
    }

    // zeros must be committed to LDS before async-DMA writes land on top of them
    asm volatile("s_wait_dscnt 0x0" ::: "memory");
    __syncthreads();

    // ---- 3) async-stage x window into LDS (CDNA5 async copy, ASYNCcnt-tracked) ----
    // element space: kh in [0,7) x b in [0,BT) x c in [0,NCOL)
    for (int idx = tid; idx < KHN * BT * NCOL; idx += 64) {
        int kh  = idx / (BT * NCOL);
        int rem = idx - kh * (BT * NCOL);
        int b   = rem / NCOL;
        int c   = rem - b * NCOL;
        int r   = STR * i  - PADN + kh;    // input row
        int ca  = STR * j0 - PADN + c;     // input col
        if (r >= 0 && r < HDIM && ca >= 0 && ca < WDIM) {
            uint32_t lds    = (uint32_t)(uintptr_t)&xs[(kh * BT + b) * XSTRIDE + c];
            const float* g  = &x[((size_t)(b0 + b) * HDIM + r) * WDIM + ca];
            asm volatile("global_load_async_to_lds_b32 %0, %1, off"
                         :: "v"(lds), "v"(g) : "memory");
        }
    }
    asm volatile("s_wait_asynccnt 0x0" ::: "memory");
    __syncthreads();

    // ---- 4) banded WMMA: D(16b x 16j) += A(16b x 4k) * B(4k x 16j), f32 ----
    const int nn = lane & 15;   // A: batch-in-tile ; B: j-in-tile ; D: N index
    const int hl = lane >> 4;   // 0: K={0,1}/M=r ; 1: K={2,3}/M=8+r

    v8f acc = {0.f, 0.f, 0.f, 0.f, 0.f, 0.f, 0.f, 0.f};
    #pragma unroll
    for (int kh = 0; kh < KHN; ++kh) {
        const float* ax = &xs[(kh * BT + wv * 16 + nn) * XSTRIDE + 2 * hl];
        const float* bx = &bb[(kh * JT + nn)           * XSTRIDE + 2 * hl];
        #pragma unroll
        for (int ks = 0; ks < NK / 4; ++ks) {
            v2f a = *(const v2f*)(ax + 4 * ks);   // ds_load_b64
            v2f b = *(const v2f*)(bx + 4 * ks);   // ds_load_b64
            acc = __builtin_amdgcn_wmma_f32_16x16x4_f32(
                false, a, false, b, (short)0, acc, false, false);
        }
    }

    // ---- 5) epilogue: add bias, scatter D fragment ----
    const int j  = j0 + nn;
    const float bv = bias[i * NOUT + j];
    #pragma unroll
    for (int r = 0; r < 8; ++r) {
        int m = hl * 8 + r;                 // M index 0..15
        int b = b0 + wv * 16 + m;           // batch
        out[((size_t)b * NOUT + i) * NOUT + j] = acc[r] + bv;
    }
}

extern "C" void kernel_launch(void* const* d_in, const int* in_sizes, int n_in,
                              void* d_out, int out_size, void* d_ws, size_t ws_size,
                              hipStream_t stream) {
    (void)in_sizes; (void)n_in; (void)d_ws; (void)ws_size; (void)out_size;
    const float* x    = (const float*)d_in[0];
    const float* wgt  = (const float*)d_in[1];
    const float* bias = (const float*)d_in[2];
    float* out        = (float*)d_out;

    dim3 grid(NOUT / JT, NOUT, BATCH / BT);   // (7, 112, 2)
    dim3 block(64, 1, 1);                     // 2 wave32
    lc2d_wmma_kernel<<<grid, block, 0, stream>>>(x, wgt, bias, out);
}